// FOFE_Encoding_53601191854897
// MI455X (gfx1250) — compile-verified
//
#include <hip/hip_runtime.h>
#include <cstdint>

#define LCHARS 20
#define VOCAB 256
#define WAVES_PER_BLOCK 8

// One wave32 per output row (256 fp32 vocab counts).
// Per row:
//   1. s_wait_asynccnt 2  -- drain async stores that used this LDS buffer
//      (double buffer, 2 outstanding stores per iteration)
//   2. zero 1KB LDS buffer: 2x ds_store_b128 per lane
//   3. lanes 0..19 load their char id and do ONE ds_add_f32 scatter of
//      alpha^(19-lane) into the LDS row (LDS atomics resolve duplicate ids)
//   4. s_wait_dscnt 0  -- DS and ASYNC counters are unordered w.r.t. each
//      other; the async engine must see the completed scatter
//   5. 2x global_store_async_from_lds_b128 per lane: LDS -> HBM with no
//      VGPR round trip, tracked by ASYNCcnt
__global__ __launch_bounds__(256)
void fofe_encode_kernel(const int* __restrict__ ids,
                        const float* __restrict__ forgetting_factor,
                        float* __restrict__ out,
                        int nrows, int wave_stride)
{
    __shared__ float smem[WAVES_PER_BLOCK * 2 * VOCAB];  // 16 KB

    const int lane = threadIdx.x & 31;
    const int warp = threadIdx.x >> 5;
    const int wave_id = blockIdx.x * WAVES_PER_BLOCK + warp;

    const float alpha = forgetting_factor[0];

    // p = alpha^(19 - lane) for lanes 0..19, branch-free binary
    // exponentiation (5 square-and-select steps, no divergent loop).
    int e = (LCHARS - 1) - lane;          // may be negative for lanes >= 20
    if (e < 0) e = 0;                     // lanes 20..31 never use p anyway
    float p = 1.0f;
    float a = alpha;
#pragma unroll
    for (int b = 0; b < 5; ++b) {
        p = (e & (1 << b)) ? p * a : p;   // v_cndmask-selected multiply
        a *= a;
    }

    float* wave_base = smem + warp * (2 * VOCAB);
    // Low 32 bits of the generic pointer to a __shared__ object are the
    // workgroup-relative LDS byte offset (local->flat addrspacecast layout).
    const uint32_t wave_base_lds = (uint32_t)(uintptr_t)wave_base;

    int it = 0;
    for (int row = wave_id; row < nrows; row += wave_stride, ++it) {
        float*   buf     = wave_base + (it & 1) * VOCAB;
        uint32_t buf_lds = wave_base_lds + (uint32_t)((it & 1) * VOCAB * 4);

        // Buffer reused every 2 iterations; each iteration leaves 2 async
        // stores outstanding, so ASYNCcnt <= 2 guarantees this buffer's
        // previous stores have drained.
        asm volatile("s_wait_asynccnt 2" ::: "memory");

        // Zero the 1KB row: lane covers bytes [lane*32, lane*32+32)
        float4 z = make_float4(0.f, 0.f, 0.f, 0.f);
        float4* b4 = (float4*)buf;
        b4[lane * 2 + 0] = z;
        b4[lane * 2 + 1] = z;

        // Scatter: one ds_add_f32 wave instruction covers all 20 chars
        if (lane < LCHARS) {
            int id = ids[row * LCHARS + lane];
            atomicAdd(&buf[id], p);
        }

        // DS (zero-stores + atomic adds) must complete before the async
        // engine reads LDS; DScnt and ASYNCcnt are independent counters.
        asm volatile("s_wait_dscnt 0" ::: "memory");

        // Async LDS -> global, 1KB per row, 32B per lane (2 x b128)
        uint64_t g0 = (uint64_t)(uintptr_t)out
                    + (uint64_t)row * (VOCAB * 4)
                    + (uint32_t)(lane * 32);
        uint32_t l0 = buf_lds + (uint32_t)(lane * 32);
        asm volatile(
            "global_store_async_from_lds_b128 %0, %2, off\n\t"
            "global_store_async_from_lds_b128 %1, %3, off"
            :: "v"(g0), "v"(g0 + 16), "v"(l0), "v"(l0 + 16)
            : "memory");
    }

    // S_ENDPGM does an implicit wait-idle, but be explicit anyway.
    asm volatile("s_wait_asynccnt 0" ::: "memory");
}

extern "C" void kernel_launch(void* const* d_in, const int* in_sizes, int n_in,
                              void* d_out, int out_size, void* d_ws, size_t ws_size,
                              hipStream_t stream) {
    const int*   ids = (const int*)d_in[0];          // [B*W*L] int32
    const float* ff  = (const float*)d_in[1];        // [1] float32
    float*       out = (float*)d_out;                // [B*W*VOCAB] float32

    const int nrows  = in_sizes[0] / LCHARS;         // B*W = 131072
    const int blocks = 2048;                         // 16384 waves -> 8 rows/wave
    const int wave_stride = blocks * WAVES_PER_BLOCK;

    fofe_encode_kernel<<<blocks, 256, 0, stream>>>(ids, ff, out, nrows, wave_stride);
}